// sml_4956392259672
// MI455X (gfx1250) — compile-verified
//
#include <hip/hip_runtime.h>

#define BATCH 512
#define EMB   512
#define NCLS  100

typedef __attribute__((ext_vector_type(2))) float v2f;
typedef __attribute__((ext_vector_type(8))) float v8f;

// ---------------------------------------------------------------------------
// Kernel 1: D = X @ X^T  (512x512x512 fp32) via V_WMMA_F32_16X16X4_F32.
// One 16x16 output tile per wave; 8 waves/block; 128 blocks cover 1024 tiles.
// ---------------------------------------------------------------------------
__global__ __launch_bounds__(256) void gemm_xxt_wmma(const float* __restrict__ X,
                                                     float* __restrict__ D) {
  const int wave = threadIdx.x >> 5;
  const int lane = threadIdx.x & 31;
  const int tile = blockIdx.x * 8 + wave;      // 0..1023
  const int m0 = (tile >> 5) << 4;             // row base of tile
  const int n0 = (tile & 31) << 4;             // col base of tile
  const int r     = lane & 15;
  const int khalf = (lane >> 4) << 1;          // lanes 0-15 -> k+0, lanes 16-31 -> k+2

  // A fragment: A[m0+r][k+khalf+{0,1}]  (ISA 7.12.2 fp32 16x4 layout)
  // B fragment: B[k][n] = X[n0+n][k] -> same pattern with column base n0.
  const float* arow = X + (size_t)(m0 + r) * EMB + khalf;
  const float* brow = X + (size_t)(n0 + r) * EMB + khalf;

  v8f c = {};
#pragma unroll 4
  for (int k = 0; k < EMB; k += 4) {
    v2f a = *(const v2f*)(arow + k);
    v2f b = *(const v2f*)(brow + k);
    // (neg_a, A, neg_b, B, c_mod, C, reuse_a, reuse_b)
    c = __builtin_amdgcn_wmma_f32_16x16x4_f32(false, a, false, b, (short)0, c,
                                              false, false);
  }

  // C/D layout: VGPR q <-> row m0 + q + 8*(lane>=16), col n0 + (lane&15)
  const int rowOff = (lane >> 4) << 3;
  float* dptr = D + (size_t)(m0 + rowOff) * BATCH + n0 + r;
#pragma unroll
  for (int q = 0; q < 8; ++q) dptr[(size_t)q * BATCH] = c[q];
}

// ---------------------------------------------------------------------------
// Kernel 2: per-anchor hinge sum.  One block per anchor i.
// partial[i] = sum_{j in pos(i)} sum_{k in neg(i)} relu(D[i,j]-D[i,k]+mu_i)
// Deterministic: serial positive-list build, fixed-order accumulation,
// fixed binary-tree block reduction.
// ---------------------------------------------------------------------------
__global__ __launch_bounds__(256) void hinge_rows(const float* __restrict__ D,
                                                  const int* __restrict__ labels,
                                                  const float* __restrict__ mu,
                                                  float* __restrict__ partial) {
  __shared__ float rowD[BATCH];
  __shared__ int   lab[BATCH];
  __shared__ int   posIdx[BATCH];
  __shared__ int   nPosS;
  __shared__ float red[256];

  const int i = blockIdx.x;
  const int t = threadIdx.x;

  for (int j = t; j < BATCH; j += 256) {
    rowD[j] = D[(size_t)i * BATCH + j];
    lab[j]  = labels[j];
  }
  __syncthreads();

  const int li = lab[i];
  if (t == 0) {
    int np = 0;
    for (int j = 0; j < BATCH; ++j)
      if (lab[j] == li && j != i) posIdx[np++] = j;
    nPosS = np;
  }
  __syncthreads();

  const int   np  = nPosS;
  const float mui = mu[li];

  float acc = 0.0f;
  for (int k = t; k < BATCH; k += 256) {
    if (lab[k] != li) {                      // valid negative
      const float base = mui - rowD[k];
      for (int p = 0; p < np; ++p) {
        const float v = rowD[posIdx[p]] + base;
        acc += fmaxf(v, 0.0f);
      }
    }
  }

  red[t] = acc;
  __syncthreads();
#pragma unroll
  for (int s = 128; s > 0; s >>= 1) {
    if (t < s) red[t] += red[t + s];
    __syncthreads();
  }
  if (t == 0) partial[i] = red[0];
}

// ---------------------------------------------------------------------------
// Kernel 3: final scalar = sum(partial) - sum(mu)/C - sum(nv)/C
// ---------------------------------------------------------------------------
__global__ __launch_bounds__(256) void finalize_loss(const float* __restrict__ partial,
                                                     const float* __restrict__ mu,
                                                     const float* __restrict__ nv,
                                                     float* __restrict__ out) {
  __shared__ float red[256];
  const int t = threadIdx.x;
  red[t] = partial[t] + partial[t + 256];
  __syncthreads();
#pragma unroll
  for (int s = 128; s > 0; s >>= 1) {
    if (t < s) red[t] += red[t + s];
    __syncthreads();
  }
  if (t == 0) {
    float sm = 0.0f, sn = 0.0f;
    for (int c = 0; c < NCLS; ++c) { sm += mu[c]; sn += nv[c]; }
    out[0] = red[0] - sm / (float)NCLS - sn / (float)NCLS;
  }
}

// ---------------------------------------------------------------------------
extern "C" void kernel_launch(void* const* d_in, const int* in_sizes, int n_in,
                              void* d_out, int out_size, void* d_ws, size_t ws_size,
                              hipStream_t stream) {
  const float* X      = (const float*)d_in[0];   // [512, 512] f32
  const int*   labels = (const int*)d_in[1];     // [512] int
  const float* mu     = (const float*)d_in[2];   // [100] f32
  const float* nv     = (const float*)d_in[3];   // [100] f32

  float* D       = (float*)d_ws;                 // 512*512 floats = 1 MB
  float* partial = D + (size_t)BATCH * BATCH;    // 512 floats

  gemm_xxt_wmma<<<128, 256, 0, stream>>>(X, D);
  hinge_rows<<<BATCH, 256, 0, stream>>>(D, labels, mu, partial);
  finalize_loss<<<1, 256, 0, stream>>>(partial, mu, nv, (float*)d_out);
}